// TmixWrapper_46797963657369
// MI455X (gfx1250) — compile-verified
//
#include <hip/hip_runtime.h>
#include <stdint.h>

// ---- problem constants (match reference) ----
#define T_   2048
#define C_   2048
#define M_   32
#define D_   64
#define KQK  1024
#define KV   1024

typedef float    v2f  __attribute__((ext_vector_type(2)));
typedef float    v8f  __attribute__((ext_vector_type(8)));
typedef uint32_t u32x4 __attribute__((ext_vector_type(4)));
typedef int      i32x4 __attribute__((ext_vector_type(4)));
typedef int      i32x8 __attribute__((ext_vector_type(8)));

// -----------------------------------------------------------------------------
// TDM: 1-D DMA of 1024 f32 (4 KB, one weight column-tile) global -> LDS.
// Descriptor per CDNA5 ISA ch.8: group0 {count=1, lds_addr, global_addr, type=2},
// group1 {data_size=4B, tensor_dim0=tile_dim0=1024, dim1=1, stride0=1024}.
// -----------------------------------------------------------------------------
__device__ __forceinline__ void tdm_load_4kb(const float* gsrc, float* lds_dst) {
  uint64_t ga = (uint64_t)(uintptr_t)gsrc;
  uint32_t la = (uint32_t)(uintptr_t)lds_dst;     // flat->LDS: low 32 bits
  u32x4 g0;
  g0[0] = 1u;                                     // count=1, user mode
  g0[1] = la;                                     // lds_addr
  g0[2] = (uint32_t)ga;                           // global_addr[31:0]
  g0[3] = (uint32_t)(ga >> 32) | (2u << 30);      // global_addr[56:32] | type=2
  i32x8 g1;
  g1[0] = (int)(2u << 16);                        // data_size = 4B, wg_mask=0
  g1[1] = (int)(1024u << 16);                     // tensor_dim0 = 1024 (lo16)
  g1[2] = (int)(1u << 16);                        // dim0 hi=0, tensor_dim1 = 1
  g1[3] = (int)(1024u << 16);                     // tile_dim0 = 1024
  g1[4] = 1;                                      // tile_dim1 = 1
  g1[5] = 1024;                                   // tensor_dim0_stride = 1024
  g1[6] = 0;
  g1[7] = 0;
  i32x4 g2 = {0, 0, 0, 0};
  i32x4 g3 = {0, 0, 0, 0};
#if defined(__clang_major__) && (__clang_major__ >= 23)
  i32x8 g2b = {0, 0, 0, 0, 0, 0, 0, 0};
  __builtin_amdgcn_tensor_load_to_lds(g0, g1, g2, g3, g2b, 0);
#else
  __builtin_amdgcn_tensor_load_to_lds(g0, g1, g2, g3, 0);
#endif
}

// -----------------------------------------------------------------------------
// Kernel 1: fused per-head GEMM (x_m[16,64] x w_m^T[64,1024]) + row argmax.
// One workgroup = 128 rows of one head, one projection. 8 waves x 16 rows.
// Weight tiles double-buffered in LDS via the Tensor Data Mover: DMA of
// tile n+1 overlaps the 16 chained v_wmma_f32_16x16x4_f32 of tile n.
// One barrier per iteration.
// -----------------------------------------------------------------------------
__global__ __launch_bounds__(256) void route_argmax_kernel(
    const float* __restrict__ x,
    const float* __restrict__ wq,
    const float* __restrict__ wk,
    const float* __restrict__ wv,
    int* __restrict__ routes)   // [3][M][T]
{
  __shared__ float bt[2][16 * D_];           // 2 x 4 KB weight tiles

  const int m    = blockIdx.y;
  const int pr   = blockIdx.z;
  const float* w = (pr == 0) ? wq : (pr == 1) ? wk : wv;
  int* route     = routes + (size_t)pr * (M_ * T_) + m * T_;

  const int wave = threadIdx.x >> 5;
  const int lane = threadIdx.x & 31;
  const int half = lane >> 4;                // 0: rows 0-7 / K lo, 1: rows 8-15 / K hi
  const int l16  = lane & 15;
  const int t0   = blockIdx.x * 128 + wave * 16;

  // ---- preload A fragments: 16 chunks covering the D=64 contraction ----
  // 32-bit A 16x4 layout: lanes 0-15 hold K=4j..4j+1, lanes 16-31 hold 4j+2..4j+3
  v2f a[16];
  {
    const float* xrow = x + (size_t)(t0 + l16) * C_ + m * D_;
    #pragma unroll
    for (int j = 0; j < 16; ++j) {
      a[j] = *reinterpret_cast<const v2f*>(xrow + 4 * j + half * 2);
    }
  }

  float bestv[8];
  int   besti[8];
  #pragma unroll
  for (int r = 0; r < 8; ++r) { bestv[r] = -3.4e38f; besti[r] = 0; }

  const float* wbase = w + (size_t)m * KQK * D_;

  // ---- prologue: DMA tile 0 ----
  if (threadIdx.x < 32) {
    tdm_load_4kb(wbase, &bt[0][0]);
    __builtin_amdgcn_s_wait_tensorcnt(0);
  }
  __syncthreads();

  int buf = 0;
  for (int n0 = 0; n0 < KQK; n0 += 16, buf ^= 1) {
    // ---- issue DMA for tile n0+16 into the other buffer (overlaps compute) ----
    if (n0 + 16 < KQK && threadIdx.x < 32) {
      tdm_load_4kb(wbase + (size_t)(n0 + 16) * D_, &bt[buf ^ 1][0]);
    }

    // ---- batch all B fragments (one DS wait), then 16 back-to-back WMMAs ----
    const float* cur = &bt[buf][0];
    v2f b[16];
    #pragma unroll
    for (int j = 0; j < 16; ++j) {
      // 32-bit B 4x16 layout mirrors A: lane = N, VGPR pair = K chunk
      b[j] = *reinterpret_cast<const v2f*>(cur + l16 * D_ + 4 * j + half * 2);
    }
    v8f acc = {};
    #pragma unroll
    for (int j = 0; j < 16; ++j) {
      acc = __builtin_amdgcn_wmma_f32_16x16x4_f32(
          /*neg_a=*/false, a[j], /*neg_b=*/false, b[j],
          /*c_mod=*/(short)0, acc, /*reuse_a=*/false, /*reuse_b=*/false);
    }

    // ---- fused running argmax (strict '>' keeps earliest index) ----
    const int col = n0 + l16;
    #pragma unroll
    for (int r = 0; r < 8; ++r) {
      if (acc[r] > bestv[r]) { bestv[r] = acc[r]; besti[r] = col; }
    }

    // ---- single barrier per iteration: next tile landed, current tile free ----
    if (threadIdx.x < 32) __builtin_amdgcn_s_wait_tensorcnt(0);
    __syncthreads();
  }

  // ---- reduce across the 16 column-lanes of each half ----
  #pragma unroll
  for (int r = 0; r < 8; ++r) {
    float v = bestv[r]; int i = besti[r];
    #pragma unroll
    for (int off = 8; off >= 1; off >>= 1) {
      float ov = __shfl_xor(v, off, 16);
      int   oi = __shfl_xor(i, off, 16);
      if (ov > v || (ov == v && oi < i)) { v = ov; i = oi; }
    }
    bestv[r] = v; besti[r] = i;
  }
  if (l16 == 0) {                             // lane 0 -> rows 0-7, lane 16 -> rows 8-15
    #pragma unroll
    for (int r = 0; r < 8; ++r) route[t0 + half * 8 + r] = besti[r];
  }
}

// -----------------------------------------------------------------------------
// Kernel 2: per-head sequential last-occurrence scan. One wave per head.
// last[] table and v-ids live in LDS; wave-uniform DS ops are in-order.
// -----------------------------------------------------------------------------
__global__ __launch_bounds__(32) void scan_kernel(const int* __restrict__ routes,
                                                  int* __restrict__ idx_out)
{
  __shared__ int last[KQK];
  __shared__ int vids[T_];
  const int m    = blockIdx.x;
  const int lane = threadIdx.x;
  const int* rq  = routes + 0 * (M_ * T_) + m * T_;
  const int* rk  = routes + 1 * (M_ * T_) + m * T_;
  const int* rv  = routes + 2 * (M_ * T_) + m * T_;

  for (int i = lane; i < KQK; i += 32) last[i] = -1;
  for (int i = lane; i < T_;  i += 32) vids[i] = rv[i];
  __syncthreads();

  int* out = idx_out + m * T_;
  for (int t0 = 0; t0 < T_; t0 += 32) {
    const int q = rq[t0 + lane];
    const int k = rk[t0 + lane];
    int my = 0;
    #pragma unroll 4
    for (int i = 0; i < 32; ++i) {
      const int qi  = __shfl(q, i);
      const int ki  = __shfl(k, i);
      const int tau = last[qi];               // read-before-write: in-order DS
      last[ki] = t0 + i;
      const int val = (tau >= 0) ? (vids[tau] + 1) : 0;
      if (lane == i) my = val;
    }
    out[t0 + lane] = my;
  }
}

// -----------------------------------------------------------------------------
// Kernel 3: embedding gather, float4-vectorized. idx==0 -> zero (padding row).
// -----------------------------------------------------------------------------
__global__ __launch_bounds__(256) void gather_kernel(const float* __restrict__ emb,
                                                     const int* __restrict__ idx,
                                                     float* __restrict__ out)
{
  const int gid = blockIdx.x * 256 + threadIdx.x;  // one float4 per thread
  const int t   = gid >> 9;                        // 512 float4 per row
  const int c4  = gid & 511;
  const int m   = c4 >> 4;                         // 16 float4 per route part
  const int d4  = (c4 & 15) << 2;
  const int id  = idx[m * T_ + t];
  float4 r = make_float4(0.f, 0.f, 0.f, 0.f);
  if (id > 0) {
    const float* e = emb + ((size_t)m * (KV + 1) + id) * D_ + d4;
    r = *reinterpret_cast<const float4*>(e);
  }
  reinterpret_cast<float4*>(out)[gid] = r;
}

// -----------------------------------------------------------------------------
extern "C" void kernel_launch(void* const* d_in, const int* in_sizes, int n_in,
                              void* d_out, int out_size, void* d_ws, size_t ws_size,
                              hipStream_t stream) {
  const float* x   = (const float*)d_in[0];
  const float* wq  = (const float*)d_in[1];
  const float* wk  = (const float*)d_in[2];
  const float* wv  = (const float*)d_in[3];
  const float* emb = (const float*)d_in[4];

  int* routes = (int*)d_ws;                 // [3][M][T]  = 768 KB
  int* idx    = routes + 3 * M_ * T_;       // [M][T]     = 256 KB

  dim3 g1(T_ / 128, M_, 3);
  route_argmax_kernel<<<g1, 256, 0, stream>>>(x, wq, wk, wv, routes);
  scan_kernel<<<M_, 32, 0, stream>>>(routes, idx);
  gather_kernel<<<(T_ * C_ / 4) / 256, 256, 0, stream>>>(emb, idx, (float*)d_out);
}